// PCENAdaptation_EMA_21844203667490
// MI455X (gfx1250) — compile-verified
//
#include <hip/hip_runtime.h>
#include <stdint.h>

// PCEN with EMA over time on MI455X (gfx1250).
// x: (32,128,8000) fp32 -> 4096 independent channels of T=8000.
// One wave32 per channel; lane owns 2 consecutive timesteps (pair-folded scan).
// Data path: global_load_async_to_lds_b64 ring buffer (depth 16), ASYNCcnt waits.

#define T_LEN 8000
#define CHUNK 64                  // floats per async op per wave (2 floats / lane)
#define NITER (T_LEN / CHUNK)     // 125
#define NBUF 16                   // ring depth: 16 x 256B = 4KB in flight per wave
#define WAVES_PER_BLOCK 8
#define THREADS (WAVES_PER_BLOCK * 32)

typedef float v2f __attribute__((ext_vector_type(2)));

__device__ __forceinline__ float fast_log2(float v) { return __builtin_amdgcn_logf(v); }   // v_log_f32
__device__ __forceinline__ float fast_exp2(float v) { return __builtin_amdgcn_exp2f(v); }  // v_exp_f32

__device__ __forceinline__ void async_load_b64(uint32_t lds_byte_addr,
                                               unsigned long long gaddr) {
    // VDST = LDS byte address VGPR, VADDR = 64-bit global address VGPR pair
    asm volatile("global_load_async_to_lds_b64 %0, %1, off"
                 :
                 : "v"(lds_byte_addr), "v"(gaddr)
                 : "memory");
}

__device__ __forceinline__ void wait_asynccnt_15() {
    asm volatile("s_wait_asynccnt 0xf" ::: "memory");
}
__device__ __forceinline__ void wait_asynccnt_0() {
    asm volatile("s_wait_asynccnt 0x0" ::: "memory");
}

__global__ __launch_bounds__(THREADS) void pcen_ema_kernel(
    const float* __restrict__ x,
    const float* __restrict__ alphaP,
    const float* __restrict__ deltaP,
    const float* __restrict__ rP,
    float* __restrict__ y)
{
    __shared__ __align__(16) float sbuf[WAVES_PER_BLOCK][NBUF][CHUNK]; // 32 KB

    const int wave = threadIdx.x >> 5;
    const int lane = threadIdx.x & 31;
    const int chan = blockIdx.x * WAVES_PER_BLOCK + wave;

    const float* xc = x + (size_t)chan * T_LEN;
    float*       yc = y + (size_t)chan * T_LEN;

    const float alpha = alphaP[0];
    const float delta = deltaP[0];
    const float r     = rP[0];

    // delta^r, computed once
    const float pd = fast_exp2(r * fast_log2(delta));

    // EMA constants: M_t = a*M_{t-1} + s*x_t, a = 1 - SMOOTH
    constexpr float S    = 0.025f;
    constexpr float A1   = 0.975f;
    constexpr float INVA = 1.0f / 0.975f;
    constexpr float P1   = A1 * A1;          // a^2  (pair-domain step 1)
    constexpr float P2   = P1 * P1;          // a^4
    constexpr float P4   = P2 * P2;          // a^8
    constexpr float P8   = P4 * P4;          // a^16
    constexpr float P16  = P8 * P8;          // a^32
    constexpr float EPSC = 1e-6f;

    // per-lane carry decay a^(2*lane+2); log2(0.975) = -0.03652578
    const float dpl = fast_exp2((float)(2 * lane + 2) * -0.03652578f);

    // LDS byte addresses for the async engine (low 32 bits of flat LDS addr)
    const uint32_t ldsBase  = (uint32_t)(uintptr_t)(&sbuf[wave][0][0]);
    const uint32_t laneSlot = (uint32_t)lane * 8u;

    // ---- prologue: fill the async pipeline ----
    #pragma unroll
    for (int i = 0; i < NBUF; ++i) {
        async_load_b64(ldsBase + (uint32_t)i * (CHUNK * 4) + laneSlot,
                       (unsigned long long)(uintptr_t)(xc + i * CHUNK + lane * 2));
    }

    float carry = 0.0f;

    auto process = [&](int i) {
        // lane reads back exactly the 8B slot the async engine wrote: t = i*64 + 2*lane (+1)
        const v2f xv2 = *(const v2f*)&sbuf[wave][i & (NBUF - 1)][lane * 2];
        const float x0 = fmaxf(xv2.x, 0.0f);
        const float x1 = fmaxf(xv2.y, 0.0f);

        // pair term: z_l = s*(a*x0 + x1); scan in pair domain with decay a^2
        float z = S * fmaf(A1, x0, x1);
        float up;
        up = __shfl_up(z, 1,  32); if (lane >= 1)  z = fmaf(P1,  up, z);
        up = __shfl_up(z, 2,  32); if (lane >= 2)  z = fmaf(P2,  up, z);
        up = __shfl_up(z, 4,  32); if (lane >= 4)  z = fmaf(P4,  up, z);
        up = __shfl_up(z, 8,  32); if (lane >= 8)  z = fmaf(P8,  up, z);
        up = __shfl_up(z, 16, 32); if (lane >= 16) z = fmaf(P16, up, z);

        // M at odd t = i*64 + 2*lane + 1, including carry: + carry * a^(2l+2)
        const float Modd = fmaf(carry, dpl, z);
        carry = __shfl(Modd, 31, 32);

        // exact reconstruction of even t: M_{2l} = (M_{2l+1} - s*x1) / a
        const float Meven = (Modd - S * x1) * INVA;

        // y = (x * (eps+M)^(-alpha) + delta)^r - delta^r
        const float n0 = x0 * fast_exp2(-alpha * fast_log2(EPSC + Meven));
        const float n1 = x1 * fast_exp2(-alpha * fast_log2(EPSC + Modd));
        v2f o;
        o.x = fast_exp2(r * fast_log2(n0 + delta)) - pd;
        o.y = fast_exp2(r * fast_log2(n1 + delta)) - pd;

        __builtin_nontemporal_store(o, (v2f*)&yc[i * CHUNK + lane * 2]);
    };

    // ---- steady state: wait oldest, process, refill ----
    for (int i = 0; i < NITER - NBUF; ++i) {
        wait_asynccnt_15();
        process(i);
        const int j = i + NBUF;
        async_load_b64(ldsBase + (uint32_t)(j & (NBUF - 1)) * (CHUNK * 4) + laneSlot,
                       (unsigned long long)(uintptr_t)(xc + j * CHUNK + lane * 2));
    }

    // ---- drain ----
    wait_asynccnt_0();
    for (int i = NITER - NBUF; i < NITER; ++i) {
        process(i);
    }
}

extern "C" void kernel_launch(void* const* d_in, const int* in_sizes, int n_in,
                              void* d_out, int out_size, void* d_ws, size_t ws_size,
                              hipStream_t stream) {
    const float* x      = (const float*)d_in[0];
    const float* alphaP = (const float*)d_in[1];
    const float* deltaP = (const float*)d_in[2];
    const float* rP     = (const float*)d_in[3];
    float* y = (float*)d_out;

    const int nchan = in_sizes[0] / T_LEN;           // 32*128 = 4096
    const int grid  = nchan / WAVES_PER_BLOCK;       // 512 blocks x 256 threads
    pcen_ema_kernel<<<grid, THREADS, 0, stream>>>(x, alphaP, deltaP, rP, y);
}